// EGAT_26482768347461
// MI455X (gfx1250) — compile-verified
//
#include <hip/hip_runtime.h>
#include <hip/hip_bf16.h>

typedef __attribute__((ext_vector_type(2))) float v2f;
typedef __attribute__((ext_vector_type(8))) float v8f;
typedef __attribute__((ext_vector_type(4))) unsigned u32x4;
typedef __attribute__((ext_vector_type(8))) int i32x8;
typedef __attribute__((ext_vector_type(4))) int i32x4;

#define NB   16
#define NPGc 1024
#define DINc 16
#define EDc  8
#define HEADSc 5
#define OUT1c  6
#define HIDc  30
#define C1c   32
#define C2c   4
#define NT   (NB*NPGc)     /* 16384 nodes */
#define ET   (NT*32)       /* 524288 edges */

// ---------------- workspace layout (float units) ----------------
// zero-initialized region first
constexpr size_t OFF_MENC  = 0;                                   // uint, N*5
constexpr size_t OFF_DEN1  = OFF_MENC  + (size_t)NT*5;
constexpr size_t OFF_X1    = OFF_DEN1  + (size_t)NT*5;            // N*30
constexpr size_t OFF_MENC2 = OFF_X1    + (size_t)NT*30;           // uint, N
constexpr size_t OFF_DEN2  = OFF_MENC2 + (size_t)NT;
constexpr size_t OFF_S1    = OFF_DEN2  + (size_t)NT;              // N*32
constexpr size_t OFF_STAT  = OFF_S1    + (size_t)NT*32;           // 8: adjsum,trace,gfro,...
constexpr size_t ZTOT      = OFF_STAT + 8;
// non-zeroed region
constexpr size_t OFF_H     = ZTOT;                                // N*30
constexpr size_t OFF_HS    = OFF_H    + (size_t)NT*30;            // N*5
constexpr size_t OFF_HD    = OFF_HS   + (size_t)NT*5;             // N*5
constexpr size_t OFF_LG    = OFF_HD   + (size_t)NT*5;             // 5*E
constexpr size_t OFF_HP    = OFF_LG   + (size_t)5*ET;             // N*32
constexpr size_t OFF_HPS   = OFF_HP   + (size_t)NT*32;            // N
constexpr size_t OFF_HPD   = OFF_HPS  + (size_t)NT;               // N
constexpr size_t OFF_LG2   = OFF_HPD  + (size_t)NT;               // E
constexpr size_t OFF_SS    = OFF_LG2  + (size_t)ET;               // N*32 softmaxed s
constexpr size_t OFF_T     = OFF_SS   + (size_t)NT*32;            // B*1024*32  (adj @ s)
constexpr size_t OFF_ADJ2  = OFF_T    + (size_t)NT*32;            // 16*32*32
constexpr size_t OFF_X2    = OFF_ADJ2 + (size_t)NB*C1c*C1c;       // 512*30
constexpr size_t OFF_HP2   = OFF_X2   + (size_t)NB*C1c*HIDc;      // 512*4
constexpr size_t OFF_HPS2  = OFF_HP2  + (size_t)NB*C1c*C2c;      // 512
constexpr size_t OFF_HPD2  = OFF_HPS2 + (size_t)NB*C1c;           // 512
constexpr size_t OFF_S2    = OFF_HPD2 + (size_t)NB*C1c;           // 512*4
constexpr size_t WS_TOT    = OFF_S2   + (size_t)NB*C1c*C2c;       // ~6.66M floats (~27MB)

// monotonic float<->uint encoding for atomicMax on floats (init 0 == "-inf")
__device__ __forceinline__ unsigned fenc(float f) {
  unsigned u = __float_as_uint(f);
  return (u & 0x80000000u) ? ~u : (u | 0x80000000u);
}
__device__ __forceinline__ float fdec(unsigned u) {
  return __uint_as_float((u & 0x80000000u) ? (u & 0x7FFFFFFFu) : ~u);
}
__device__ __forceinline__ float lrelu(float v) { return v >= 0.f ? v : 0.2f * v; }

// TDM: async-load a 64x64 fp32 tile (row stride 1024 floats in memory) into LDS,
// padding +4 DWORDs every 64 DWORDs so the LDS row stride is 68 floats
// (bank-conflict-free for the WMMA fragment reads).
__device__ __forceinline__ void tdm_load_tile64(const float* gtile, unsigned lds_byte) {
  unsigned long long ga = (unsigned long long)(size_t)gtile;
  u32x4 g0; i32x8 g1; i32x4 gz4; i32x8 gz8;
  g0[0] = 1u;                                             // count=1, user descriptor
  g0[1] = lds_byte;                                       // lds_addr
  g0[2] = (unsigned)(ga & 0xFFFFFFFFull);                 // global_addr lo
  g0[3] = (unsigned)((ga >> 32) & 0x1FFFFFFull) | (2u << 30); // global_addr hi | type=2
  g1[0] = (int)((2u << 16)    // data_size = 4 bytes
              | (1u << 20)    // pad_enable
              | (5u << 22)    // pad_interval: every 64 DWORDs
              | (3u << 25));  // pad_amount: 4 DWORDs
  g1[1] = (int)(1024u << 16);  // [47:32] barrier addr=0 | [63:48] tensor_dim0 lo16
  g1[2] = (int)(1024u << 16);  // [79:64] tensor_dim0 hi16=0 | [95:80] tensor_dim1 lo16
  g1[3] = (int)(64u << 16);    // [111:96] tensor_dim1 hi16=0 | [127:112] tile_dim0=64
  g1[4] = 64;                  // [143:128] tile_dim1=64 | [159:144] tile_dim2=0
  g1[5] = 1024;                // tensor_dim0_stride lo32
  g1[6] = 0;                   // stride hi | tensor_dim1_stride lo
  g1[7] = 0;
  gz4[0] = 0; gz4[1] = 0; gz4[2] = 0; gz4[3] = 0;         // 2D: groups 2/3 unused
  for (int i = 0; i < 8; i++) gz8[i] = 0;
  __builtin_amdgcn_tensor_load_to_lds(g0, g1, gz4, gz4, gz8, 0);
}

// ---------------- kernels ----------------
__global__ void k_init(float* ws, size_t n) {
  size_t i = (size_t)blockIdx.x * 256 + threadIdx.x;
  if (i < n) ws[i] = 0.f;
}

// h = x @ W1 ; hs/hd = per-head attention partials
__global__ void k_feat1(const float* __restrict__ x, const float* __restrict__ W1,
                        const float* __restrict__ as1, const float* __restrict__ ad1,
                        float* __restrict__ h, float* __restrict__ hs, float* __restrict__ hd) {
  __shared__ float sW[DINc * HIDc], sAs[HEADSc * OUT1c], sAd[HEADSc * OUT1c];
  int t = threadIdx.x;
  for (int i = t; i < DINc * HIDc; i += 256) sW[i] = W1[i];
  for (int i = t; i < HEADSc * OUT1c; i += 256) { sAs[i] = as1[i]; sAd[i] = ad1[i]; }
  __syncthreads();
  int n = blockIdx.x * 256 + t;
  float xr[DINc];
  for (int i = 0; i < DINc; i++) xr[i] = x[(size_t)n * DINc + i];
  float hr[HIDc];
  for (int f = 0; f < HIDc; f++) {
    float a = 0.f;
    for (int i = 0; i < DINc; i++) a += xr[i] * sW[i * HIDc + f];
    hr[f] = a; h[(size_t)n * HIDc + f] = a;
  }
  for (int k = 0; k < HEADSc; k++) {
    float s = 0.f, d = 0.f;
    for (int c = 0; c < OUT1c; c++) {
      s += hr[k * OUT1c + c] * sAs[k * OUT1c + c];
      d += hr[k * OUT1c + c] * sAd[k * OUT1c + c];
    }
    hs[(size_t)n * 5 + k] = s; hd[(size_t)n * 5 + k] = d;
  }
}

__global__ void k_logit1(const int* __restrict__ src, const int* __restrict__ dst,
                         const float* __restrict__ ea, const float* __restrict__ We,
                         const float* __restrict__ hs, const float* __restrict__ hd,
                         float* __restrict__ lg, unsigned* __restrict__ menc) {
  int e = blockIdx.x * 256 + threadIdx.x;
  int s = src[e], d = dst[e];
  float a[EDc];
  for (int j = 0; j < EDc; j++) a[j] = ea[(size_t)e * EDc + j];
  for (int k = 0; k < HEADSc; k++) {
    float v = hs[(size_t)s * 5 + k] + hd[(size_t)d * 5 + k];
    for (int j = 0; j < EDc; j++) v += a[j] * We[j * HEADSc + k];
    v = lrelu(v);
    lg[(size_t)k * ET + e] = v;
    atomicMax(&menc[(size_t)d * 5 + k], fenc(v));
  }
}

__global__ void k_exp1(const int* __restrict__ dst, float* __restrict__ lg,
                       const unsigned* __restrict__ menc, float* __restrict__ den) {
  int e = blockIdx.x * 256 + threadIdx.x;
  int d = dst[e];
  for (int k = 0; k < HEADSc; k++) {
    float m = fdec(menc[(size_t)d * 5 + k]);
    float ex = expf(lg[(size_t)k * ET + e] - m);
    lg[(size_t)k * ET + e] = ex;
    atomicAdd(&den[(size_t)d * 5 + k], ex);
  }
}

__global__ void k_acc1(const int* __restrict__ src, const int* __restrict__ dst,
                       const float* __restrict__ lg, const float* __restrict__ den,
                       const float* __restrict__ h, float* __restrict__ x1) {
  int e = blockIdx.x * 256 + threadIdx.x;
  int s = src[e], d = dst[e];
  for (int k = 0; k < HEADSc; k++) {
    float alpha = lg[(size_t)k * ET + e] / (den[(size_t)d * 5 + k] + 1e-16f);
    for (int c = 0; c < OUT1c; c++) {
      int f = k * OUT1c + c;
      atomicAdd(&x1[(size_t)d * HIDc + f], alpha * h[(size_t)s * HIDc + f]);
    }
  }
}

// hp = x1 @ Wp1 (N x 32); hps/hpd partials (1 head)
__global__ void k_feat2(const float* __restrict__ x1, const float* __restrict__ Wp1,
                        const float* __restrict__ asp, const float* __restrict__ adp,
                        float* __restrict__ hp, float* __restrict__ hps, float* __restrict__ hpd) {
  __shared__ float sW[HIDc * C1c], sA[C1c], sD[C1c];
  int t = threadIdx.x;
  for (int i = t; i < HIDc * C1c; i += 256) sW[i] = Wp1[i];
  if (t < C1c) { sA[t] = asp[t]; sD[t] = adp[t]; }
  __syncthreads();
  int n = blockIdx.x * 256 + t;
  float xr[HIDc];
  for (int f = 0; f < HIDc; f++) xr[f] = x1[(size_t)n * HIDc + f];
  float ps = 0.f, pd = 0.f;
  for (int c = 0; c < C1c; c++) {
    float a = 0.f;
    for (int f = 0; f < HIDc; f++) a += xr[f] * sW[f * C1c + c];
    hp[(size_t)n * C1c + c] = a;
    ps += a * sA[c]; pd += a * sD[c];
  }
  hps[n] = ps; hpd[n] = pd;
}

__global__ void k_logit2(const int* __restrict__ src, const int* __restrict__ dst,
                         const float* __restrict__ ea, const float* __restrict__ Wep1,
                         const float* __restrict__ hps, const float* __restrict__ hpd,
                         float* __restrict__ lg2, unsigned* __restrict__ menc2) {
  int e = blockIdx.x * 256 + threadIdx.x;
  int s = src[e], d = dst[e];
  float v = hps[s] + hpd[d];
  for (int j = 0; j < EDc; j++) v += ea[(size_t)e * EDc + j] * Wep1[j];
  v = lrelu(v);
  lg2[e] = v;
  atomicMax(&menc2[d], fenc(v));
}

__global__ void k_exp2(const int* __restrict__ dst, float* __restrict__ lg2,
                       const unsigned* __restrict__ menc2, float* __restrict__ den2) {
  int e = blockIdx.x * 256 + threadIdx.x;
  int d = dst[e];
  float ex = expf(lg2[e] - fdec(menc2[d]));
  lg2[e] = ex;
  atomicAdd(&den2[d], ex);
}

__global__ void k_acc2(const int* __restrict__ src, const int* __restrict__ dst,
                       const float* __restrict__ lg2, const float* __restrict__ den2,
                       const float* __restrict__ hp, float* __restrict__ s1) {
  int e = blockIdx.x * 256 + threadIdx.x;
  int s = src[e], d = dst[e];
  float alpha = lg2[e] / (den2[d] + 1e-16f);
  for (int c = 0; c < C1c; c++)
    atomicAdd(&s1[(size_t)d * C1c + c], alpha * hp[(size_t)s * C1c + c]);
}

__global__ void k_smax(const float* __restrict__ s1, float* __restrict__ sS) {
  int n = blockIdx.x * 256 + threadIdx.x;
  float r[C1c];
  float m = -1e30f;
  for (int c = 0; c < C1c; c++) { r[c] = s1[(size_t)n * C1c + c]; m = fmaxf(m, r[c]); }
  float sum = 0.f;
  for (int c = 0; c < C1c; c++) { r[c] = expf(r[c] - m); sum += r[c]; }
  float inv = 1.f / sum;
  for (int c = 0; c < C1c; c++) sS[(size_t)n * C1c + c] = r[c] * inv;
}

// T[b] = adj[b] @ s[b] via V_WMMA_F32_16X16X4_F32.
// adj tiles streamed by TDM (tensor_load_to_lds) with LDS padding, double-buffered;
// sum(adj) accumulated from A-fragments of the nsub==0 waves.
// block = 256 (8 waves); block tile: 64 rows x 32 cols; grid (16 row-tiles, 16 batches)
__global__ __launch_bounds__(256) void k_adjs(const float* __restrict__ adj,
                                              const float* __restrict__ sS,
                                              float* __restrict__ T,
                                              float* __restrict__ stats) {
  __shared__ float aT[2][64 * 68];   // adj tile, TDM-padded stride 68
  __shared__ float sT[2][64 * 33];   // s tile, padded stride
  __shared__ float redSum;
  const int b = blockIdx.y;
  const int rowBase = blockIdx.x * 64;
  const int t = threadIdx.x;
  const int lane = t & 31, wave = t >> 5;
  const int msub = wave >> 1, nsub = wave & 1;
  const int half = lane >> 4, l = lane & 15;
  const float* adjTile = adj + (size_t)b * NPGc * NPGc + (size_t)rowBase * NPGc;

  v8f acc;
  for (int i = 0; i < 8; i++) acc[i] = 0.f;
  float asum = 0.f;

  // prologue: kick off chunk 0
  if (wave == 0) tdm_load_tile64(adjTile, (unsigned)(size_t)&aT[0][0]);
  for (int it = 0; it < 2; it++) {
    int q = t + it * 256;
    int r = q >> 3, c4 = (q & 7) * 4;
    const float* p = sS + ((size_t)(b << 10) + r) * C1c + c4;
    float* dp = &sT[0][r * 33 + c4];
    dp[0] = p[0]; dp[1] = p[1]; dp[2] = p[2]; dp[3] = p[3];
  }

  for (int c = 0; c < 16; c++) {
    const int cur = c & 1;
    if (c + 1 < 16) {
      // prefetch next chunk into the other buffers
      if (wave == 0)
        tdm_load_tile64(adjTile + (size_t)(c + 1) * 64, (unsigned)(size_t)&aT[cur ^ 1][0]);
      int kcn = (c + 1) * 64;
      for (int it = 0; it < 2; it++) {
        int q = t + it * 256;
        int r = q >> 3, c4 = (q & 7) * 4;
        const float* p = sS + ((size_t)(b << 10) + kcn + r) * C1c + c4;
        float* dp = &sT[cur ^ 1][r * 33 + c4];
        dp[0] = p[0]; dp[1] = p[1]; dp[2] = p[2]; dp[3] = p[3];
      }
    }
    if (wave == 0) {
      if (c + 1 < 16) __builtin_amdgcn_s_wait_tensorcnt(1);  // current chunk landed
      else            __builtin_amdgcn_s_wait_tensorcnt(0);  // last chunk landed
    }
    __syncthreads();
    for (int kk = 0; kk < 64; kk += 4) {
      v2f af, bf;
      af[0] = aT[cur][(msub * 16 + l) * 68 + kk + 2 * half + 0];
      af[1] = aT[cur][(msub * 16 + l) * 68 + kk + 2 * half + 1];
      bf[0] = sT[cur][(kk + 0 + 2 * half) * 33 + nsub * 16 + l];
      bf[1] = sT[cur][(kk + 1 + 2 * half) * 33 + nsub * 16 + l];
      if (nsub == 0) asum += af[0] + af[1];  // wave-uniform branch: each adj elem once
      acc = __builtin_amdgcn_wmma_f32_16x16x4_f32(false, af, false, bf,
                                                  (short)0, acc, false, false);
    }
    __syncthreads();
  }
  for (int r = 0; r < 8; r++) {
    int row = rowBase + msub * 16 + r + 8 * half;
    T[((size_t)(b << 10) + row) * C1c + nsub * 16 + l] = acc[r];
  }
  if (t == 0) redSum = 0.f;
  __syncthreads();
  atomicAdd(&redSum, asum);
  __syncthreads();
  if (t == 0) atomicAdd(&stats[0], redSum);
}

// per batch: out[c][j] = sum_k s[k][c] * V[k][j], V = [T | x1 | s]  (j: 0..93)
// -> adj2 (j<32), x2 (32..61), Gram Frobenius + trace accumulators
__global__ __launch_bounds__(256) void k_pool1(const float* __restrict__ sS,
                                               const float* __restrict__ T,
                                               const float* __restrict__ x1,
                                               float* __restrict__ adj2,
                                               float* __restrict__ x2,
                                               float* __restrict__ stats) {
  __shared__ float sB[128 * 33];
  __shared__ float vB[128 * 97];
  const int b = blockIdx.x;
  const int t = threadIdx.x;
  float acc[12]; int ci[12], ji[12];
  for (int o = 0; o < 12; o++) {
    int idx = t + o * 256;
    acc[o] = 0.f; ci[o] = idx / 94; ji[o] = idx % 94;
  }
  for (int kc = 0; kc < NPGc; kc += 128) {
    for (int q = t; q < 128 * 32; q += 256) {
      int r = q >> 5, c = q & 31;
      sB[r * 33 + c] = sS[((size_t)(b << 10) + kc + r) * C1c + c];
    }
    for (int q = t; q < 128 * 94; q += 256) {
      int r = q / 94, j = q % 94;
      size_t node = (size_t)(b << 10) + kc + r;
      float v;
      if (j < 32) v = T[node * C1c + j];
      else if (j < 62) v = x1[node * HIDc + (j - 32)];
      else v = sS[node * C1c + (j - 62)];
      vB[r * 97 + j] = v;
    }
    __syncthreads();
    for (int kk = 0; kk < 128; kk++) {
      const float* srow = &sB[kk * 33];
      const float* vrow = &vB[kk * 97];
#pragma unroll
      for (int o = 0; o < 12; o++) acc[o] += srow[ci[o]] * vrow[ji[o]];
    }
    __syncthreads();
  }
  for (int o = 0; o < 12; o++) {
    int idx = t + o * 256;
    if (idx >= 32 * 94) break;
    int c = ci[o], j = ji[o];
    float v = acc[o];
    if (j < 32) {
      adj2[(size_t)b * 1024 + c * 32 + j] = v;
      if (j == c) atomicAdd(&stats[1], v);        // trace(adjp)
    } else if (j < 62) {
      x2[((size_t)b * C1c + c) * HIDc + (j - 32)] = v;
    } else {
      atomicAdd(&stats[2], v * v);                // ||s^T s||_F^2
    }
  }
}

__global__ void k_feat3(const float* __restrict__ x2, const float* __restrict__ Wp2,
                        const float* __restrict__ asp2, const float* __restrict__ adp2,
                        float* __restrict__ hp2, float* __restrict__ hps2, float* __restrict__ hpd2) {
  int n = blockIdx.x * 256 + threadIdx.x;
  if (n >= NB * C1c) return;
  float ps = 0.f, pd = 0.f;
  for (int c = 0; c < C2c; c++) {
    float a = 0.f;
    for (int f = 0; f < HIDc; f++) a += x2[(size_t)n * HIDc + f] * Wp2[f * C2c + c];
    hp2[(size_t)n * C2c + c] = a;
    ps += a * asp2[c]; pd += a * adp2[c];
  }
  hps2[n] = ps; hpd2[n] = pd;
}

// dense pooled EGAT: one wave per (batch, dst); lanes = src nodes
__global__ __launch_bounds__(256) void k_s2(const float* __restrict__ hp2,
                                            const float* __restrict__ hps2,
                                            const float* __restrict__ hpd2,
                                            const float* __restrict__ adj2,
                                            const float* __restrict__ Wep2,
                                            float* __restrict__ s2) {
  int t = threadIdx.x;
  int lane = t & 31, wave = t >> 5;
  int g = blockIdx.x * 8 + wave;          // 0..511
  int b = g >> 5, j = g & 31, i = lane;
  float we = Wep2[0];
  int srcn = b * C1c + i;
  float lg = hps2[srcn] + hpd2[b * C1c + j] + adj2[(size_t)b * 1024 + i * 32 + j] * we;
  lg = lrelu(lg);
  float m = lg;
  for (int o = 16; o >= 1; o >>= 1) m = fmaxf(m, __shfl_xor(m, o, 32));
  float ex = expf(lg - m);
  float den = ex;
  for (int o = 16; o >= 1; o >>= 1) den += __shfl_xor(den, o, 32);
  float alpha = ex / (den + 1e-16f);
  for (int c = 0; c < C2c; c++) {
    float v = alpha * hp2[(size_t)srcn * C2c + c];
    for (int o = 16; o >= 1; o >>= 1) v += __shfl_xor(v, o, 32);
    if (lane == 0) s2[((size_t)b * C1c + j) * C2c + c] = v;
  }
}

// softmax(s2) -> x3 -> reg2 -> MLP head -> logits, reg
__global__ __launch_bounds__(512) void k_final(const float* __restrict__ s2,
                                               const float* __restrict__ x2,
                                               const float* __restrict__ adj2,
                                               const float* __restrict__ Wf1, const float* __restrict__ bf1,
                                               const float* __restrict__ Wf2, const float* __restrict__ bf2,
                                               const float* __restrict__ stats,
                                               float* __restrict__ out) {
  __shared__ float s2s[512 * 4];
  __shared__ float x3[NB * C2c * HIDc];   // 1920
  __shared__ float zL[512];
  __shared__ float r2red;
  int t = threadIdx.x;
  if (t == 0) r2red = 0.f;
  {
    float r[4]; float m = -1e30f;
    for (int c = 0; c < 4; c++) { r[c] = s2[(size_t)t * 4 + c]; m = fmaxf(m, r[c]); }
    float sum = 0.f;
    for (int c = 0; c < 4; c++) { r[c] = expf(r[c] - m); sum += r[c]; }
    float inv = 1.f / sum;
    for (int c = 0; c < 4; c++) s2s[t * 4 + c] = r[c] * inv;
  }
  __syncthreads();
  for (int idx = t; idx < NB * C2c * HIDc; idx += 512) {
    int b = idx / 120, rem = idx % 120, c2 = rem / 30, d = rem % 30;
    float a = 0.f;
    for (int n = 0; n < 32; n++)
      a += s2s[(b * 32 + n) * 4 + c2] * x2[((size_t)b * 32 + n) * HIDc + d];
    x3[idx] = a;
  }
  float racc = 0.f;
  for (int p = t; p < NB * 1024; p += 512) {
    int b = p >> 10, r = (p >> 5) & 31, cc = p & 31;
    float dp = 0.f;
    for (int c = 0; c < 4; c++)
      dp += s2s[(b * 32 + r) * 4 + c] * s2s[(b * 32 + cc) * 4 + c];
    float df = adj2[(size_t)b * 1024 + r * 32 + cc] - dp;
    racc += df * df;
  }
  atomicAdd(&r2red, racc);
  __syncthreads();
  {
    int b = t >> 5, o = t & 31;
    float a = bf1[o];
    for (int f = 0; f < 120; f++) a += x3[b * 120 + f] * Wf1[f * 32 + o];
    zL[t] = fmaxf(a, 0.f);
  }
  __syncthreads();
  if (t < 32) {
    int b = t >> 1, o = t & 1;
    float a = bf2[o];
    for (int f = 0; f < 32; f++) a += zL[b * 32 + f] * Wf2[f * 2 + o];
    out[b * 2 + o] = a;
  }
  if (t == 0) {
    float reg1 = (stats[0] - 2.f * stats[1] + stats[2]) / 16777216.f;  // B*NPG*NPG
    float reg2 = r2red / 16384.f;                                      // B*C1*C1
    out[32] = reg1 * 10.f + reg2 * 0.1f;
  }
}

extern "C" void kernel_launch(void* const* d_in, const int* in_sizes, int n_in,
                              void* d_out, int out_size, void* d_ws, size_t ws_size,
                              hipStream_t stream) {
  const float* x    = (const float*)d_in[0];
  const int*   ei   = (const int*)d_in[1];
  const float* ea   = (const float*)d_in[2];
  const float* adj  = (const float*)d_in[4];
  const float* W1   = (const float*)d_in[5];
  const float* as1  = (const float*)d_in[6];
  const float* ad1  = (const float*)d_in[7];
  const float* We1  = (const float*)d_in[8];
  const float* Wp1  = (const float*)d_in[9];
  const float* asp1 = (const float*)d_in[10];
  const float* adp1 = (const float*)d_in[11];
  const float* Wep1 = (const float*)d_in[12];
  const float* Wp2  = (const float*)d_in[13];
  const float* asp2 = (const float*)d_in[14];
  const float* adp2 = (const float*)d_in[15];
  const float* Wep2 = (const float*)d_in[16];
  const float* Wf1  = (const float*)d_in[17];
  const float* bf1  = (const float*)d_in[18];
  const float* Wf2  = (const float*)d_in[19];
  const float* bf2  = (const float*)d_in[20];

  const int* src = ei;
  const int* dst = ei + ET;

  float* ws = (float*)d_ws;                       // needs WS_TOT*4 ~ 27MB
  unsigned* menc  = (unsigned*)(ws + OFF_MENC);
  float*    den1  = ws + OFF_DEN1;
  float*    x1    = ws + OFF_X1;
  unsigned* menc2 = (unsigned*)(ws + OFF_MENC2);
  float*    den2  = ws + OFF_DEN2;
  float*    s1    = ws + OFF_S1;
  float*    stats = ws + OFF_STAT;
  float*    h     = ws + OFF_H;
  float*    hs    = ws + OFF_HS;
  float*    hd    = ws + OFF_HD;
  float*    lg    = ws + OFF_LG;
  float*    hp    = ws + OFF_HP;
  float*    hps   = ws + OFF_HPS;
  float*    hpd   = ws + OFF_HPD;
  float*    lg2   = ws + OFF_LG2;
  float*    sS    = ws + OFF_SS;
  float*    T     = ws + OFF_T;
  float*    adj2  = ws + OFF_ADJ2;
  float*    x2    = ws + OFF_X2;
  float*    hp2   = ws + OFF_HP2;
  float*    hps2  = ws + OFF_HPS2;
  float*    hpd2  = ws + OFF_HPD2;
  float*    s2    = ws + OFF_S2;

  k_init<<<(unsigned)((ZTOT + 255) / 256), 256, 0, stream>>>(ws, ZTOT);

  k_feat1 <<<NT / 256, 256, 0, stream>>>(x, W1, as1, ad1, h, hs, hd);
  k_logit1<<<ET / 256, 256, 0, stream>>>(src, dst, ea, We1, hs, hd, lg, menc);
  k_exp1  <<<ET / 256, 256, 0, stream>>>(dst, lg, menc, den1);
  k_acc1  <<<ET / 256, 256, 0, stream>>>(src, dst, lg, den1, h, x1);

  k_feat2 <<<NT / 256, 256, 0, stream>>>(x1, Wp1, asp1, adp1, hp, hps, hpd);
  k_logit2<<<ET / 256, 256, 0, stream>>>(src, dst, ea, Wep1, hps, hpd, lg2, menc2);
  k_exp2  <<<ET / 256, 256, 0, stream>>>(dst, lg2, menc2, den2);
  k_acc2  <<<ET / 256, 256, 0, stream>>>(src, dst, lg2, den2, hp, s1);

  k_smax  <<<NT / 256, 256, 0, stream>>>(s1, sS);
  k_adjs  <<<dim3(NPGc / 64, NB), 256, 0, stream>>>(adj, sS, T, stats);
  k_pool1 <<<NB, 256, 0, stream>>>(sS, T, x1, adj2, x2, stats);

  k_feat3 <<<2, 256, 0, stream>>>(x2, Wp2, asp2, adp2, hp2, hps2, hpd2);
  k_s2    <<<64, 256, 0, stream>>>(hp2, hps2, hpd2, adj2, Wep2, s2);
  k_final <<<1, 512, 0, stream>>>(s2, x2, adj2, Wf1, bf1, Wf2, bf2, stats, (float*)d_out);
}